// AdaptiveFieldEvolution_52810917871692
// MI455X (gfx1250) — compile-verified
//
#include <hip/hip_runtime.h>
#include <math.h>

// ---------------------------------------------------------------------------
// AdaptiveFieldEvolution on MI455X (gfx1250, wave32)
// fp32 WMMA (V_WMMA_F32_16X16X4_F32) for the two per-pixel 1x1-conv GEMMs.
// ---------------------------------------------------------------------------

typedef __attribute__((ext_vector_type(2))) float v2f;
typedef __attribute__((ext_vector_type(8))) float v8f;

constexpr int   kB    = 4;
constexpr int   kC    = 64;     // channels
constexpr int   kC2   = 128;    // hidden channels
constexpr int   kH    = 128;
constexpr int   kW    = 128;
constexpr int   kNPIX = kH * kW;                 // per-channel plane
constexpr long  kNTOT = (long)kB * kC * kH * kW; // 4,194,304 elements
constexpr int   kSteps = 50;
constexpr float kDT = 0.1f;
constexpr float kThresh = 0.01f;
constexpr int   kBlocks = kB * kH;               // 512 row-blocks
constexpr int   kPitch = kW + 4;                 // LDS pad: bank-conflict-free

// ---------------------------------------------------------------------------
// prep: transpose weights into k-major layout + zero control block
//   w1T[c*kC2 + o] = w1[o*kC + c]      (64 x 128)
//   w2T[k*kC  + o] = w2[o*kC2 + k]     (128 x 64)
// ---------------------------------------------------------------------------
__global__ __launch_bounds__(256)
void prep_kernel(const float* __restrict__ w1, const float* __restrict__ w2,
                 float* __restrict__ w1T, float* __restrict__ w2T,
                 int* __restrict__ ctrl) {
  int gid = blockIdx.x * 256 + threadIdx.x;     // 0 .. 16383
  if (gid < kC * kC2) {                         // w1T
    int o = gid % kC2, c = gid / kC2;
    w1T[c * kC2 + o] = w1[o * kC + c];
  } else {                                      // w2T
    int i = gid - kC * kC2;
    int o = i % kC, k = i / kC;
    w2T[k * kC + o] = w2[o * kC2 + k];
  }
  if (gid == 0) { ctrl[0] = 0; ctrl[1] = 0; }   // done, steps
}

// ---------------------------------------------------------------------------
// one Euler step: fused depthwise-conv + MLP(1x1 convs via fp32 WMMA)
// grid = 512 (one workgroup per (b,h) row), block = 256 (8 wave32)
// ---------------------------------------------------------------------------
__global__ __launch_bounds__(256)
void step_kernel(const float* __restrict__ fin, float* __restrict__ fout,
                 const float* __restrict__ w1T, const float* __restrict__ w2T,
                 const float* __restrict__ dw,  const float* __restrict__ db,
                 const float* __restrict__ b1,  const float* __restrict__ b2,
                 const float* __restrict__ dcoeff,
                 float* __restrict__ partial, const int* __restrict__ ctrl) {
  __shared__ float sF[kC][kW];        // this row, all channels   (32 KB)
  __shared__ float sH[kC2][kPitch];   // gelu(H), then react      (66 KB)
  __shared__ float sRed[256];

  const int tid = threadIdx.x;
  const int b   = blockIdx.x >> 7;
  const int h   = blockIdx.x & 127;
  const float* frow = fin  + ((size_t)b * kC) * kNPIX + (size_t)h * kW;
  float*       orow = fout + ((size_t)b * kC) * kNPIX + (size_t)h * kW;

  if (ctrl[0]) {                      // converged: freeze (copy row)
    for (int i = tid; i < kC * kW; i += 256) {
      int c = i >> 7, x = i & 127;
      orow[(size_t)c * kNPIX + x] = frow[(size_t)c * kNPIX + x];
    }
    return;
  }

  // ---- Phase 1: stage field row (64 ch x 128 px) into LDS, float4 ----------
  for (int i = tid; i < (kC * kW) / 4; i += 256) {
    int c = i >> 5, x4 = i & 31;                         // 32 float4 per row
    float4 v = ((const float4*)(frow + (size_t)c * kNPIX))[x4];
    ((float4*)&sF[c][0])[x4] = v;
  }
  __syncthreads();

  const int wav   = tid >> 5;
  const int lane  = tid & 31;
  const int lhalf = lane >> 4;        // K sub-block select for A/B frags
  const int l16   = lane & 15;
  const int pixA  = wav * 16 + l16;   // A-matrix M index (pixel)

  const v8f vzero = {0.f, 0.f, 0.f, 0.f, 0.f, 0.f, 0.f, 0.f};

  // ---- Phase 2: GEMM1  H = F * W1^T  (M=16 px/wave, N=128, K=64) ----------
  v8f acc1[8];
#pragma unroll
  for (int nt = 0; nt < 8; ++nt) acc1[nt] = vzero;

#pragma unroll 4
  for (int k0 = 0; k0 < kC; k0 += 4) {
    const int ka = k0 + 2 * lhalf;
    v2f a;
    a.x = sF[ka][pixA];
    a.y = sF[ka + 1][pixA];
#pragma unroll
    for (int nt = 0; nt < 8; ++nt) {
      v2f bf;
      bf.x = w1T[ka * kC2 + nt * 16 + l16];
      bf.y = w1T[(ka + 1) * kC2 + nt * 16 + l16];
      acc1[nt] = __builtin_amdgcn_wmma_f32_16x16x4_f32(
          false, a, false, bf, (short)0, acc1[nt], false, false);
    }
  }

  // bias + exact GELU, scatter to sH[o][pix]
#pragma unroll
  for (int nt = 0; nt < 8; ++nt) {
    const int o = nt * 16 + l16;
    const float bias = b1[o];
#pragma unroll
    for (int r = 0; r < 8; ++r) {
      float x = acc1[nt][r] + bias;
      float g = 0.5f * x * (1.0f + erff(x * 0.70710678118654752f));
      sH[o][wav * 16 + lhalf * 8 + r] = g;
    }
  }
  __syncthreads();

  // ---- Phase 3: GEMM2  react = gelu(H) * W2^T  (N=64, K=128) --------------
  v8f acc2[4];
#pragma unroll
  for (int nt = 0; nt < 4; ++nt) acc2[nt] = vzero;

#pragma unroll 4
  for (int k0 = 0; k0 < kC2; k0 += 4) {
    const int ka = k0 + 2 * lhalf;
    v2f a;
    a.x = sH[ka][pixA];
    a.y = sH[ka + 1][pixA];
#pragma unroll
    for (int nt = 0; nt < 4; ++nt) {
      v2f bf;
      bf.x = w2T[ka * kC + nt * 16 + l16];
      bf.y = w2T[(ka + 1) * kC + nt * 16 + l16];
      acc2[nt] = __builtin_amdgcn_wmma_f32_16x16x4_f32(
          false, a, false, bf, (short)0, acc2[nt], false, false);
    }
  }
  __syncthreads();                    // done reading sH as H

  // stage react into sH rows [0,64)
#pragma unroll
  for (int nt = 0; nt < 4; ++nt) {
    const int o = nt * 16 + l16;
#pragma unroll
    for (int r = 0; r < 8; ++r)
      sH[o][wav * 16 + lhalf * 8 + r] = acc2[nt][r];
  }
  __syncthreads();

  // ---- Phase 4: depthwise 3x3 + Euler update + |delta| reduction ----------
  const float dc = dcoeff[0];
  float lsum = 0.0f;
  for (int it = 0; it < 32; ++it) {
    const int idx = it * 256 + tid;
    const int c = idx >> 7, x = idx & 127;
    const float fold = sF[c][x];
    const float* wc = dw + c * 9;

    float diff;
    {
      float l = (x > 0)   ? sF[c][x - 1] : 0.f;
      float r = (x < 127) ? sF[c][x + 1] : 0.f;
      diff = wc[3] * l + wc[4] * fold + wc[5] * r;
    }
    if (h > 0) {
      const float* g = frow + (size_t)c * kNPIX - kW;
      float m  = g[x];
      float ml = (x > 0)   ? g[x - 1] : 0.f;
      float mr = (x < 127) ? g[x + 1] : 0.f;
      diff += wc[0] * ml + wc[1] * m + wc[2] * mr;
    }
    if (h < 127) {
      const float* g = frow + (size_t)c * kNPIX + kW;
      float m  = g[x];
      float ml = (x > 0)   ? g[x - 1] : 0.f;
      float mr = (x < 127) ? g[x + 1] : 0.f;
      diff += wc[6] * ml + wc[7] * m + wc[8] * mr;
    }
    diff += db[c];

    const float react = sH[c][x] + b2[c];
    const float nv = fold + kDT * (dc * diff + react);
    orow[(size_t)c * kNPIX + x] = nv;
    lsum += fabsf(nv - fold);
  }

  sRed[tid] = lsum;
  __syncthreads();
#pragma unroll
  for (int s = 128; s > 0; s >>= 1) {
    if (tid < s) sRed[tid] += sRed[tid + s];
    __syncthreads();
  }
  if (tid == 0) partial[blockIdx.x] = sRed[0];
}

// ---------------------------------------------------------------------------
// per-step reduction: mean|delta| -> done/steps (mirrors lax.scan body)
// ---------------------------------------------------------------------------
__global__ __launch_bounds__(512)
void reduce_kernel(const float* __restrict__ partial, int* __restrict__ ctrl) {
  __shared__ float s[512];
  const int t = threadIdx.x;
  if (ctrl[0]) return;                // already converged: steps frozen
  s[t] = partial[t];
  __syncthreads();
#pragma unroll
  for (int st = 256; st > 0; st >>= 1) {
    if (t < st) s[t] += s[t + st];
    __syncthreads();
  }
  if (t == 0) {
    ctrl[1] += 1;                                   // steps++
    if (s[0] / (float)kNTOT < kThresh) ctrl[0] = 1; // done
  }
}

__global__ void finalize_kernel(const int* __restrict__ ctrl,
                                float* __restrict__ out_steps) {
  *out_steps = (float)ctrl[1];
}

// ---------------------------------------------------------------------------
extern "C" void kernel_launch(void* const* d_in, const int* in_sizes, int n_in,
                              void* d_out, int out_size, void* d_ws, size_t ws_size,
                              hipStream_t stream) {
  const float* field  = (const float*)d_in[0];
  const float* dw     = (const float*)d_in[1];
  const float* db     = (const float*)d_in[2];
  const float* w1     = (const float*)d_in[3];
  const float* b1     = (const float*)d_in[4];
  const float* w2     = (const float*)d_in[5];
  const float* b2     = (const float*)d_in[6];
  const float* dcoeff = (const float*)d_in[7];
  // d_in[8] = max_steps (device); reference fixes it at 50 -> compile-time.

  float* ws      = (float*)d_ws;
  float* P0      = ws;                               // alt field buffer (16 MB)
  float* w1T     = ws + kNTOT;                       // 8192 floats
  float* w2T     = w1T + kC * kC2;                   // 8192 floats
  float* partial = w2T + kC2 * kC;                   // 512 floats
  int*   ctrl    = (int*)(partial + kBlocks);        // [done, steps]
  float* P1      = (float*)d_out;                    // final field lands here

  hipLaunchKernelGGL(prep_kernel, dim3(64), dim3(256), 0, stream,
                     w1, w2, w1T, w2T, ctrl);

  for (int i = 1; i <= kSteps; ++i) {
    const float* in  = (i == 1) ? field : ((i & 1) ? P1 : P0);
    float*       out = (i & 1) ? P0 : P1;            // step 50 (even) -> d_out
    hipLaunchKernelGGL(step_kernel, dim3(kBlocks), dim3(256), 0, stream,
                       in, out, w1T, w2T, dw, db, b1, b2, dcoeff, partial, ctrl);
    hipLaunchKernelGGL(reduce_kernel, dim3(1), dim3(512), 0, stream,
                       partial, ctrl);
  }

  hipLaunchKernelGGL(finalize_kernel, dim3(1), dim3(1), 0, stream,
                     ctrl, (float*)d_out + kNTOT);
}